// RV2BEV_36996848287965
// MI455X (gfx1250) — compile-verified
//
#include <hip/hip_runtime.h>
#include <cstdint>

// ---------------------------------------------------------------------------
// RV (range view) -> BEV scatter-max.
//   feat: [8, 32, 64, 2048] f32, rbin: [8, 1, 64, 2048] i32
//   out : [8, 32, 512, 512] f32
// Pipeline:
//   (1) init out keys to 0 (order-preserving uint key space; 0 < all finite)
//   (2) scatter: TDM (tensor_load_to_lds) stages the 32x256 feature tile
//       (one descriptor per block, issued by wave 0, TENSORcnt-tracked),
//       overlapped with polar->cartesian index math; then one branchless
//       global_atomic_max_u32 per channel on the key-transformed value
//   (3) finalize: decode keys in place, empty cells (key==0) -> 0.0f
// ---------------------------------------------------------------------------

#define HR   64
#define WR   2048
#define CCH  32
#define HB   512
#define WB   512
#define TILE_W 256

// linspace / geometry constants (double-accurate, stored as f32)
#define THETA_MAX_F  0.05235987755982988f    //  +3 deg
#define DTHETA_F    (-0.007757018897752576f) // (-25-3)deg / 63
#define PHI_MIN_F   (-3.14159265358979323f)  // -pi
#define DPHI_F       0.0030694603357985275f  // 2*pi / 2047
#define R_SLOPE_F    1.6f                    // (50-2)/30
#define R_BASE_F     2.8f                    // 2 + 1.6/2

typedef uint32_t v4u __attribute__((ext_vector_type(4)));
typedef int      v8i __attribute__((ext_vector_type(8)));
typedef int      v4i __attribute__((ext_vector_type(4)));

// order-preserving float -> uint key: unsigned compare == float compare
__device__ __forceinline__ uint32_t fkey(float f) {
    uint32_t u = __float_as_uint(f);
    uint32_t mask = (uint32_t)((int32_t)u >> 31) | 0x80000000u;
    return u ^ mask;   // pos: u|0x80000000 ; neg: ~u
}

__device__ __forceinline__ float funkey(uint32_t k) {
    // k MSB set (was >=0): clear MSB ; k MSB clear (was <0): bitwise-not
    uint32_t mask = 0x80000000u | (uint32_t)(~((int32_t)k >> 31));
    return __uint_as_float(k ^ mask);
}

__global__ __launch_bounds__(256) void rv2bev_init(uint4* __restrict__ out) {
    size_t i = (size_t)blockIdx.x * 256 + threadIdx.x;
    out[i] = make_uint4(0u, 0u, 0u, 0u);
}

__global__ __launch_bounds__(256) void rv2bev_scatter(
        const float* __restrict__ feat,
        const int*   __restrict__ rbin,
        uint32_t*    __restrict__ out) {
    const int t  = threadIdx.x;
    const int w0 = blockIdx.x * TILE_W;
    const int h  = blockIdx.y;
    const int b  = blockIdx.z;

    // LDS tile: [c][pixel], 32 x 256 f32 = 32 KB (well under 320 KB/WGP)
    __shared__ float sfeat[CCH * TILE_W];

    // global layout for fixed (b,h): TILE_W contiguous floats per channel,
    // channel stride = HR*WR floats -> 2D tile for the Tensor Data Mover.
    const float* gbase = feat + (((size_t)b * CCH) * HR + h) * WR + w0;
    const uint32_t lds_base = (uint32_t)(uintptr_t)(&sfeat[0]); // AS3 offset

#if __has_builtin(__builtin_amdgcn_tensor_load_to_lds)
    if (t < 32) {   // one wave issues the TDM descriptor (uniform operands)
        uint64_t ga = (uint64_t)(uintptr_t)gbase;
        v4u g0;
        g0[0] = 1u;                                   // count=1, user mode
        g0[1] = lds_base;                             // lds_addr (bytes)
        g0[2] = (uint32_t)ga;                         // global_addr[31:0]
        g0[3] = (uint32_t)((ga >> 32) & 0x01FFFFFFu)  // global_addr[56:32]
              | (2u << 30);                           // type=2 ("image")
        v8i g1;
        g1[0] = (int)(2u << 16);                      // data_size=2 -> 4 B
        g1[1] = (int)((uint32_t)TILE_W << 16);        // tensor_dim0[15:0] @63:48
        g1[2] = (int)((uint32_t)CCH    << 16);        // tensor_dim1[15:0] @95:80
        g1[3] = (int)((uint32_t)TILE_W << 16);        // tile_dim0 @127:112
        g1[4] = (int)CCH;                             // tile_dim1 @143:128
        g1[5] = (int)(HR * WR);                       // dim0_stride[31:0] = 131072
        g1[6] = 0;                                    // dim0_stride[47:32], dim1_stride lo
        g1[7] = 0;
        v4i z4 = {0, 0, 0, 0};                        // groups 2/3 unused (2D)
#if __clang_major__ >= 23
        v8i z8 = {0, 0, 0, 0, 0, 0, 0, 0};
        __builtin_amdgcn_tensor_load_to_lds(g0, g1, z4, z4, z8, 0);
#else
        __builtin_amdgcn_tensor_load_to_lds(g0, g1, z4, z4, 0);
#endif
        __builtin_amdgcn_s_wait_tensorcnt(0);         // wave 0 sees completion
    }
#else
    // fallback: per-wave async DMA copies (ASYNCcnt)
    #pragma unroll
    for (int k = 0; k < 8; ++k) {
        uint32_t q   = (uint32_t)(k * 256 + t) * 16u;
        uint32_t c   = q >> 10;
        uint32_t off = q & 1023u;
        uint64_t ga  = (uint64_t)(uintptr_t)((const char*)(gbase + (size_t)c * (HR * WR)) + off);
        uint32_t la  = lds_base + q;
        asm volatile("global_load_async_to_lds_b128 %0, %1, off"
                     :: "v"(la), "v"(ga) : "memory");
    }
    asm volatile("s_wait_asynccnt 0x0" ::: "memory");
#endif

    // ---- overlap DMA with index math (hardware trig: args are bounded) ----
    const int w   = w0 + t;
    const int bin = rbin[((size_t)b * HR + h) * WR + w];

    float r     = fmaf((float)bin, R_SLOPE_F, R_BASE_F);
    float theta = fmaf((float)h, DTHETA_F, THETA_MAX_F);   // block-uniform
    float phi   = fmaf((float)w, DPHI_F, PHI_MIN_F);       // |phi| <= pi

    float rct = r * __cosf(theta);
    float x   = rct * __cosf(phi);
    float y   = rct * __sinf(phi);

    // truncation toward zero matches jnp float->int32 cast
    int u = (int)((x + 50.0f) / 100.0f * 511.0f);
    int v = (int)((50.0f - y) / 100.0f * 511.0f);
    bool valid = (u >= 0) && (u < WB) && (v >= 0) && (v < HB);

    // publish the LDS tile (issuing wave has already waited on TENSORcnt)
    __syncthreads();

    if (valid) {
        uint32_t* obase = out + ((size_t)b * CCH) * (HB * WB) + v * WB + u;
        #pragma unroll
        for (int c = 0; c < CCH; ++c) {
            uint32_t key = fkey(sfeat[c * TILE_W + t]);
            atomicMax(obase + c * (HB * WB), key);   // global_atomic_max_u32
        }
    }
}

__global__ __launch_bounds__(256) void rv2bev_finalize(uint4* __restrict__ out) {
    size_t i = (size_t)blockIdx.x * 256 + threadIdx.x;
    uint4 k = out[i];
    float4 v;
    v.x = (k.x == 0u) ? 0.0f : funkey(k.x);
    v.y = (k.y == 0u) ? 0.0f : funkey(k.y);
    v.z = (k.z == 0u) ? 0.0f : funkey(k.z);
    v.w = (k.w == 0u) ? 0.0f : funkey(k.w);
    ((float4*)out)[i] = v;
}

extern "C" void kernel_launch(void* const* d_in, const int* in_sizes, int n_in,
                              void* d_out, int out_size, void* d_ws, size_t ws_size,
                              hipStream_t stream) {
    const float* feat = (const float*)d_in[0];
    const int*   rbin = (const int*)d_in[1];

    // out_size = 8*32*512*512 = 67,108,864 elems = 16,777,216 x uint4
    const int n4 = out_size / 4;
    rv2bev_init<<<n4 / 256, 256, 0, stream>>>((uint4*)d_out);

    dim3 grid(WR / TILE_W, HR, 8);
    rv2bev_scatter<<<grid, 256, 0, stream>>>(feat, rbin, (uint32_t*)d_out);

    rv2bev_finalize<<<n4 / 256, 256, 0, stream>>>((uint4*)d_out);
}